// MOLELayer_46677704573585
// MI455X (gfx1250) — compile-verified
//
#include <hip/hip_runtime.h>
#include <math.h>

typedef __attribute__((ext_vector_type(2))) float v2f;
typedef __attribute__((ext_vector_type(8))) float v8f;

#define BATCH 4
#define SEQ   4096
#define DIM   2048
#define NEXP  8
#define RANK  16
#define NTOK  (BATCH * SEQ)   // 16384 tokens
#define WPB   4               // waves per block
#define TPB   (WPB * 32)

// One wave handles one 16-token tile end-to-end with fp32 WMMA (16x16x4).
__global__ __launch_bounds__(TPB) void mole_kernel(
    const float* __restrict__ x,    // [T, D]
    const float* __restrict__ gW,   // [E, D]
    const float* __restrict__ gb,   // [E]
    const float* __restrict__ A,    // [E, R, D]
    const float* __restrict__ Bm,   // [E, D, R]
    float* __restrict__ out,        // [T, D]
    float* __restrict__ gp)         // [T, E]
{
    __shared__ __align__(16) float lds_H[WPB][NEXP][16][RANK];
    __shared__ __align__(16) float lds_logit[WPB][16][16];
    __shared__ __align__(16) float lds_mask[WPB][16][NEXP];

    const int tid    = threadIdx.x;
    const int wslot  = tid >> 5;
    const int lane   = tid & 31;
    const int half   = lane >> 4;   // 0 or 1
    const int lrow   = lane & 15;   // row / col within 16
    const int tile   = blockIdx.x * WPB + wslot;
    const int token0 = tile * 16;

    // -------- Phase A: h_e = X(16xD) @ A_e^T  and  logits = X @ Gpad^T ------
    v8f acc[NEXP + 1];
#pragma unroll
    for (int e = 0; e <= NEXP; ++e) acc[e] = (v8f){0,0,0,0,0,0,0,0};

    // A-frag (f32 16x4): lane<16 holds K={0,1}, lane>=16 holds K={2,3}
    const float* xrow = x + (size_t)(token0 + lrow) * DIM + 2 * half;

    for (int k0 = 0; k0 < DIM; k0 += 4) {
        v2f a = *(const v2f*)(xrow + k0);
#pragma unroll
        for (int e = 0; e < NEXP; ++e) {
            // B-frag (f32 4x16): B[k][n] = A_e[n][k]; lane n reads A_e row n
            v2f b = *(const v2f*)(A + ((size_t)(e * RANK + lrow)) * DIM + k0 + 2 * half);
            acc[e] = __builtin_amdgcn_wmma_f32_16x16x4_f32(
                false, a, false, b, (short)0, acc[e], false, false);
        }
        // Gate folded in as 9th "expert": rows 0-7 = gate_W, rows 8-15 = 0
        v2f bg = *(const v2f*)(gW + (size_t)(lrow & 7) * DIM + k0 + 2 * half);
        if (lrow >= 8) bg = (v2f){0.0f, 0.0f};
        acc[NEXP] = __builtin_amdgcn_wmma_f32_16x16x4_f32(
            false, a, false, bg, (short)0, acc[NEXP], false, false);
    }

    // Stage logits to LDS. C/D layout: VGPR i, lane j -> row i + 8*(j/16), col j%16
#pragma unroll
    for (int i = 0; i < 8; ++i)
        lds_logit[wslot][i + 8 * half][lrow] = acc[NEXP][i];
    __syncthreads();

    // -------- Softmax + top-2 mask + gate_probs output (lanes 0-15) ---------
    if (lane < 16) {
        const int t = lane;
        float l[NEXP];
        float m1 = -3.4e38f, m2 = -3.4e38f;
#pragma unroll
        for (int e = 0; e < NEXP; ++e) {
            float v = lds_logit[wslot][t][e] + gb[e];
            l[e] = v;
            if (v > m1) { m2 = m1; m1 = v; }
            else if (v > m2) { m2 = v; }
        }
        float p[NEXP];
        float s = 0.0f;
#pragma unroll
        for (int e = 0; e < NEXP; ++e) { p[e] = expf(l[e] - m1); s += p[e]; }
        float inv = 1.0f / s;
#pragma unroll
        for (int e = 0; e < NEXP; ++e) {
            gp[(size_t)(token0 + t) * NEXP + e] = p[e] * inv;
            lds_mask[wslot][t][e] = (l[e] >= m2) ? 1.0f : 0.0f;
        }
    }
    __syncthreads();

    // -------- Exact GeLU + mask, stage H to LDS -----------------------------
#pragma unroll
    for (int e = 0; e < NEXP; ++e) {
#pragma unroll
        for (int i = 0; i < 8; ++i) {
            const int m = i + 8 * half;       // token row
            float h = acc[e][i];
            float g = 0.5f * h * (1.0f + erff(h * 0.7071067811865476f));
            lds_H[wslot][e][m][lrow] = g * lds_mask[wslot][m][e];
        }
    }
    __syncthreads();

    // -------- Phase B: out(16xD) = sum_e H_e(16x16) @ B2_e(16xD) ------------
    // Hoist all H fragments into registers (A-frag layout again).
    v2f hf[NEXP][4];
#pragma unroll
    for (int e = 0; e < NEXP; ++e)
#pragma unroll
        for (int r = 0; r < 4; ++r)
            hf[e][r] = *(const v2f*)&lds_H[wslot][e][lrow][4 * r + 2 * half];

    for (int d0 = 0; d0 < DIM; d0 += 16) {
        v8f o = (v8f){0,0,0,0,0,0,0,0};
#pragma unroll
        for (int e = 0; e < NEXP; ++e) {
            // B-frag: B2[r][d] = Bm[e][d][r]; lane n owns column d0+n%16
            const float* bp = Bm + ((size_t)e * DIM + d0 + lrow) * RANK + 2 * half;
#pragma unroll
            for (int r = 0; r < 4; ++r) {
                v2f b = *(const v2f*)(bp + 4 * r);
                o = __builtin_amdgcn_wmma_f32_16x16x4_f32(
                    false, hf[e][r], false, b, (short)0, o, false, false);
            }
        }
#pragma unroll
        for (int i = 0; i < 8; ++i)
            out[(size_t)(token0 + i + 8 * half) * DIM + d0 + lrow] = o[i];
    }
}

extern "C" void kernel_launch(void* const* d_in, const int* in_sizes, int n_in,
                              void* d_out, int out_size, void* d_ws, size_t ws_size,
                              hipStream_t stream) {
    const float* x  = (const float*)d_in[0];
    const float* gW = (const float*)d_in[1];
    const float* gb = (const float*)d_in[2];
    const float* A  = (const float*)d_in[3];
    const float* Bm = (const float*)d_in[4];
    float* out = (float*)d_out;
    float* gp  = out + (size_t)NTOK * DIM;   // gate_probs follows out

    const int tiles  = NTOK / 16;            // 1024 16-token tiles
    const int blocks = tiles / WPB;          // 256 blocks x 4 waves
    mole_kernel<<<blocks, TPB, 0, stream>>>(x, gW, gb, A, Bm, out, gp);
}